// GraphPooling_47708496724384
// MI455X (gfx1250) — compile-verified
//
#include <hip/hip_runtime.h>
#include <cstdint>

// ext-vector so __builtin_nontemporal_load works (HIP float4 is a struct).
typedef float v4f __attribute__((ext_vector_type(4)));

// CDNA5 native float max atomic (no-return form -> STOREcnt, stays in L2).
// No portable HIP builtin guarantees this lowering, so use the ISA mnemonic.
__device__ __forceinline__ void gmax_f32(float* addr, float v) {
    asm volatile("global_atomic_max_num_f32 %0, %1, off"
                 :
                 : "v"((unsigned long long)(uintptr_t)addr), "v"(v)
                 : "memory");
}

__global__ void segmax_init_kernel(float* __restrict__ out, int n) {
    int i = blockIdx.x * blockDim.x + threadIdx.x;
    if (i < n) out[i] = -__builtin_inff();   // identity of max; matches jax segment_max fill
}

__global__ __launch_bounds__(256, 2)
void segmax_kernel(const float* __restrict__ x,
                   const int*   __restrict__ batch,
                   float*       __restrict__ out,
                   int N) {
    const int lane = threadIdx.x & 31;
    const int wid  = (blockIdx.x * (int)blockDim.x + (int)threadIdx.x) >> 5;
    const int nw   = ((int)gridDim.x * (int)blockDim.x) >> 5;

    // Contiguous row range per wave (batch is sorted -> few segment crossings).
    long long rows  = ((long long)N + nw - 1) / nw;
    long long start = (long long)wid * rows;
    long long end   = start + rows;
    if (end > N) end = N;
    if (start >= end) return;

    const float NEG_INF = -__builtin_inff();
    v4f acc = { NEG_INF, NEG_INF, NEG_INF, NEG_INF };
    int cur = batch[start];

    for (long long tile = start; tile < end; tile += 32) {
        // One segment-id load per lane, broadcast to the wave via ds_bpermute.
        long long li = tile + lane;
        int myseg = batch[li < end ? li : (end - 1)];

        // Prefetch next tile of x (16 KB ahead); speculative, safe past end.
        __builtin_prefetch((const void*)(x + (tile + 32) * 128 + lane * 4), 0, 1);

        int m = (int)(((end - tile) < 32) ? (end - tile) : 32);
        for (int r = 0; r < m; ++r) {
            int seg = __shfl(myseg, r, 32);
            // One row per wave: 32 lanes x float4 = 512 B = global_load_b128,
            // non-temporal (x is streamed once; 512 MB >> 192 MB L2).
            const v4f* p = (const v4f*)(x + (tile + r) * 128 + lane * 4);
            v4f v = __builtin_nontemporal_load(p);

            if (seg != cur) {
                // Segment boundary (wave-uniform branch): flush running max
                // into the 2 MB L2-resident output with native f32 max atomics.
                float* o = out + (long long)cur * 128 + lane * 4;
                gmax_f32(o + 0, acc.x);
                gmax_f32(o + 1, acc.y);
                gmax_f32(o + 2, acc.z);
                gmax_f32(o + 3, acc.w);
                acc = v;          // start new segment with this row
                cur = seg;
            } else {
                acc.x = fmaxf(acc.x, v.x);
                acc.y = fmaxf(acc.y, v.y);
                acc.z = fmaxf(acc.z, v.z);
                acc.w = fmaxf(acc.w, v.w);
            }
        }
    }

    // Final flush of the wave's last (possibly shared-with-neighbor) segment.
    float* o = out + (long long)cur * 128 + lane * 4;
    gmax_f32(o + 0, acc.x);
    gmax_f32(o + 1, acc.y);
    gmax_f32(o + 2, acc.z);
    gmax_f32(o + 3, acc.w);
}

extern "C" void kernel_launch(void* const* d_in, const int* in_sizes, int n_in,
                              void* d_out, int out_size, void* d_ws, size_t ws_size,
                              hipStream_t stream) {
    const float* x     = (const float*)d_in[0];
    const int*   batch = (const int*)d_in[1];   // jax canonicalizes int64 -> int32
    float*       out   = (float*)d_out;

    const int D = 128;
    int N = in_sizes[0] / D;                    // 1,000,000 rows

    // 1) out = -inf (max identity; harness poisons d_out, so always re-init).
    {
        int t = 256;
        int b = (out_size + t - 1) / t;
        hipLaunchKernelGGL(segmax_init_kernel, dim3(b), dim3(t), 0, stream, out, out_size);
    }

    // 2) Streaming segment-max. 512 blocks x 256 thr = 4096 waves -> ~245
    //    contiguous rows/wave vs ~244 rows/segment: ~2 atomic flushes per wave.
    {
        int t = 256;
        int b = 512;
        hipLaunchKernelGGL(segmax_kernel, dim3(b), dim3(t), 0, stream, x, batch, out, N);
    }
}